// MultiHeadAttention_43980465111551
// MI455X (gfx1250) — compile-verified
//
#include <hip/hip_runtime.h>

typedef float v2f __attribute__((ext_vector_type(2)));
typedef float v8f __attribute__((ext_vector_type(8)));

#define WMMA_F32(a, b, c) \
  __builtin_amdgcn_wmma_f32_16x16x4_f32(false, (a), false, (b), (short)0, (c), false, false)

constexpr int kB = 32;    // batch
constexpr int kL = 512;   // sequence length
constexpr int kD = 512;   // model dim
constexpr int kH = 8;     // heads
constexpr int kDH = 64;   // head dim

// ---------------------------------------------------------------------------
// Kernel 1: projection  dst = X @ W^T + bias   (X:[M,512], W:[512,512] row-major)
// Block: 256 threads = 8 waves, block computes 64x64 output tile.
// Wave (w>>1) picks 16-row slab, (w&1) picks 32-col slab (two 16x16 C tiles).
// ---------------------------------------------------------------------------
__global__ __launch_bounds__(256) void qkv_proj_kernel(
    const float* __restrict__ X, const float* __restrict__ W,
    const float* __restrict__ bias, float* __restrict__ dst) {
  const int tid = threadIdx.x;
  const int lane = tid & 31;
  const int wave = tid >> 5;
  const int g = lane >> 4;     // half-wave: selects K pair
  const int l15 = lane & 15;   // M row (A) / N col (B)

  const int m0 = blockIdx.x * 64 + (wave >> 1) * 16;
  const int n0 = blockIdx.y * 64 + (wave & 1) * 32;

  const float* arow  = X + (size_t)(m0 + l15) * kD;
  const float* b0row = W + (size_t)(n0 + l15) * kD;        // B[k][n] = W[n][k]
  const float* b1row = W + (size_t)(n0 + 16 + l15) * kD;

  v8f c0 = {}; v8f c1 = {};
  for (int k = 0; k < kD; k += 4) {
    v2f a  = *(const v2f*)(arow  + k + 2 * g);
    v2f b0 = *(const v2f*)(b0row + k + 2 * g);
    v2f b1 = *(const v2f*)(b1row + k + 2 * g);
    c0 = WMMA_F32(a, b0, c0);
    c1 = WMMA_F32(a, b1, c1);
  }

  const float bias0 = bias[n0 + l15];
  const float bias1 = bias[n0 + 16 + l15];
#pragma unroll
  for (int r = 0; r < 8; ++r) {
    const int row = m0 + r + 8 * g;           // C layout: M = r + 8*halfwave
    dst[(size_t)row * kD + n0 + l15]      = c0[r] + bias0;
    dst[(size_t)row * kD + n0 + 16 + l15] = c1[r] + bias1;
  }
}

// ---------------------------------------------------------------------------
// Kernel 2: sign-based row masks. grid.y==0 -> queries->qmask, 1 -> keys->kmask
// ---------------------------------------------------------------------------
__global__ __launch_bounds__(256) void mask_kernel(
    const float* __restrict__ queries, const float* __restrict__ keys,
    float* __restrict__ qmask, float* __restrict__ kmask) {
  const int row = blockIdx.x * 256 + threadIdx.x;
  const float* src = blockIdx.y ? keys : queries;
  float* dst = blockIdx.y ? kmask : qmask;
  const float4* p = (const float4*)(src + (size_t)row * kD);
  float s = 0.0f;
  for (int i = 0; i < kD / 4; ++i) {
    float4 v = p[i];
    s += fabsf(v.x) + fabsf(v.y) + fabsf(v.z) + fabsf(v.w);
  }
  dst[row] = (s != 0.0f) ? 1.0f : 0.0f;
}

// ---------------------------------------------------------------------------
// Kernel 3: attention for one (b, h, 64-row q tile).
// Block: 256 threads = 8 waves. LDS holds full 64x512 score strip.
// ---------------------------------------------------------------------------
constexpr int SC_PITCH = 516;  // 512 + 4 pad: 16-lane column reads hit distinct banks

__global__ __launch_bounds__(256) void attn_kernel(
    const float* __restrict__ Qb, const float* __restrict__ Kb,
    const float* __restrict__ Vb, const float* __restrict__ qmask,
    const float* __restrict__ kmask, const int* __restrict__ causality,
    float* __restrict__ outp, float* __restrict__ aw) {
  __shared__ float sc[64][SC_PITCH];     // ~129 KB
  __shared__ float rscale[64];

  const int tid = threadIdx.x;
  const int lane = tid & 31;
  const int wave = tid >> 5;
  const int g = lane >> 4;
  const int l15 = lane & 15;

  const int b = blockIdx.x / kH;
  const int h = blockIdx.x % kH;
  const int q0 = blockIdx.y * 64;
  const int causal = causality[0];

  const int mtile = wave >> 1;   // 0..3 : 16-row slab of the 64-row q tile
  const int ngrp  = wave & 1;    // splits the N dimension between wave pairs

  // ---- scores = (Q . K^T) * 0.125 with masks, into LDS ----
  {
    const float* qrow =
        Qb + (size_t)(b * kL + q0 + mtile * 16 + l15) * kD + h * kDH;
    v2f afr[16];
#pragma unroll
    for (int kk = 0; kk < 16; ++kk)
      afr[kk] = *(const v2f*)(qrow + kk * 4 + 2 * g);

    for (int j = 0; j < 16; ++j) {
      const int n0 = (ngrp * 16 + j) * 16;
      const float* krow =
          Kb + (size_t)(b * kL + n0 + l15) * kD + h * kDH;
      v8f c = {};
#pragma unroll
      for (int kk = 0; kk < 16; ++kk) {
        v2f bf = *(const v2f*)(krow + kk * 4 + 2 * g);
        c = WMMA_F32(afr[kk], bf, c);
      }
      const int col = n0 + l15;
      const float kmv = kmask[(size_t)b * kL + col];
#pragma unroll
      for (int r = 0; r < 8; ++r) {
        const int rin = mtile * 16 + r + 8 * g;
        const int qg = q0 + rin;
        float s = c[r] * 0.125f;
        if (kmv == 0.0f) s = -__builtin_inff();
        if (causal && col > qg) s = -__builtin_inff();
        sc[rin][col] = s;
      }
    }
  }
  __syncthreads();

  // ---- softmax: 4 threads per row, shuffle-reduced ----
  {
    const int row = tid >> 2;
    const int quarter = tid & 3;
    const int cbeg = quarter * 128;
    float mx = -__builtin_inff();
    for (int c = 0; c < 128; ++c) mx = fmaxf(mx, sc[row][cbeg + c]);
    mx = fmaxf(mx, __shfl_xor(mx, 1));
    mx = fmaxf(mx, __shfl_xor(mx, 2));
    float sum = 0.0f;
    const bool dead = (mx == -__builtin_inff());   // fully masked row -> nan_to_num -> 0
    for (int c = 0; c < 128; ++c) {
      const float x = sc[row][cbeg + c];
      const float e = dead ? 0.0f : __expf(x - mx);
      sc[row][cbeg + c] = e;
      sum += e;
    }
    sum += __shfl_xor(sum, 1);
    sum += __shfl_xor(sum, 2);
    if (quarter == 0) {
      rscale[row] = (dead || sum <= 0.0f)
                        ? 0.0f
                        : qmask[(size_t)b * kL + q0 + row] / sum;
    }
  }
  __syncthreads();

  // ---- normalize, apply query mask, store attention weights (coalesced f4) ----
  {
    float* awbase = aw + (((size_t)(b * kH + h)) * kL + q0) * kL;
#pragma unroll
    for (int i = 0; i < 32; ++i) {
      const int idx = tid + 256 * i;
      const int row = idx >> 7;
      const int col = (idx & 127) * 4;
      const float rs = rscale[row];
      float4 v = *(float4*)&sc[row][col];
      v.x *= rs; v.y *= rs; v.z *= rs; v.w *= rs;
      *(float4*)&sc[row][col] = v;
      *(float4*)(awbase + (size_t)row * kL + col) = v;
    }
  }
  __syncthreads();

  // ---- out tile = w(64x512, LDS) @ V(512x64) ----
  v8f o0 = {}; v8f o1 = {};
  const int nA = (ngrp * 2 + 0) * 16;
  const int nBcol = (ngrp * 2 + 1) * 16;
  const float* vbase = Vb + (size_t)(b * kL) * kD + h * kDH;
  for (int kk = 0; kk < 128; ++kk) {
    const int k = kk * 4 + 2 * g;
    v2f a = *(const v2f*)&sc[mtile * 16 + l15][k];
    const float* vr = vbase + (size_t)k * kD;
    v2f bf0, bf1;
    bf0.x = vr[nA + l15];       bf0.y = vr[kD + nA + l15];
    bf1.x = vr[nBcol + l15];    bf1.y = vr[kD + nBcol + l15];
    o0 = WMMA_F32(a, bf0, o0);
    o1 = WMMA_F32(a, bf1, o1);
  }
#pragma unroll
  for (int r = 0; r < 8; ++r) {
    const int rin = mtile * 16 + r + 8 * g;
    float* orow = outp + (size_t)(b * kL + q0 + rin) * kD + h * kDH;
    orow[nA + l15] = o0[r];
    orow[nBcol + l15] = o1[r];
  }
}

// ---------------------------------------------------------------------------
extern "C" void kernel_launch(void* const* d_in, const int* in_sizes, int n_in,
                              void* d_out, int out_size, void* d_ws, size_t ws_size,
                              hipStream_t stream) {
  const float* queries = (const float*)d_in[0];
  const float* keys    = (const float*)d_in[1];
  const float* Wq      = (const float*)d_in[2];
  const float* bq      = (const float*)d_in[3];
  const float* Wk      = (const float*)d_in[4];
  const float* bk      = (const float*)d_in[5];
  const float* Wv      = (const float*)d_in[6];
  const float* bv      = (const float*)d_in[7];
  const int*   caus    = (const int*)d_in[8];

  const size_t mat = (size_t)kB * kL * kD;   // 8.4M floats each
  float* Qb = (float*)d_ws;
  float* Kb = Qb + mat;
  float* Vb = Kb + mat;
  float* qm = Vb + mat;
  float* km = qm + (size_t)kB * kL;

  float* outp = (float*)d_out;
  float* aw   = outp + mat;                  // attention_weights follow `out`

  dim3 gproj(kB * kL / 64, kD / 64);
  qkv_proj_kernel<<<gproj, 256, 0, stream>>>(queries, Wq, bq, Qb);
  qkv_proj_kernel<<<gproj, 256, 0, stream>>>(keys,    Wk, bk, Kb);
  qkv_proj_kernel<<<gproj, 256, 0, stream>>>(keys,    Wv, bv, Vb);

  mask_kernel<<<dim3(kB * kL / 256, 2), 256, 0, stream>>>(queries, keys, qm, km);

  attn_kernel<<<dim3(kB * kH, kL / 64), 256, 0, stream>>>(
      Qb, Kb, Vb, qm, km, caus, outp, aw);
}